// TransformerBlockQuantum_65481071405230
// MI455X (gfx1250) — compile-verified
//
#include <hip/hip_runtime.h>
#include <hip/hip_bf16.h>

typedef __attribute__((ext_vector_type(2))) float v2f;
typedef __attribute__((ext_vector_type(8))) float v8f;

#define NW 8
#define FFN_DIM 1024
#define WAVES_PB 8
#define TPW 16                    // tokens per wave == WMMA M tile
#define TPB (WAVES_PB * TPW)      // 128 tokens per block
#define QP 10                     // padded pitch (floats) for 8-wide tiles (40B rows, 8B aligned)
#define HP 18                     // padded pitch (floats) for 16-wide tiles (72B rows, 8B aligned, conflict-free)
#define LN_EPS 1e-5f

// Hardware transcendental: v_cos_f32 (TRANS32, co-executes with VALU).
// Arguments here are O(1) (normal data +- small angles) so no big range reduction needed.
#define COSF(v) __cosf(v)

// V_WMMA_F32_16X16X4_F32 : D(16x16 f32) = A(16x4 f32) * B(4x16 f32) + C
// A frag: lane l holds M=l&15, K = 2*(l>>4) + {0,1}  (one v2f)
// B frag: lane l holds N=l&15, K = 2*(l>>4) + {0,1}  (one v2f)
// C frag: VGPR r, lane l -> (M = r + 8*(l>>4), N = l&15)
__device__ __forceinline__ v8f wmma4(v2f a, v2f b, v8f c) {
  return __builtin_amdgcn_wmma_f32_16x16x4_f32(false, a, false, b, (short)0, c,
                                               false, false);
}

__global__ __launch_bounds__(WAVES_PB * 32) void
qtb_fused_kernel(const float* __restrict__ x,
                 const float* __restrict__ ln1g, const float* __restrict__ ln1b,
                 const float* __restrict__ ln2g, const float* __restrict__ ln2b,
                 const float* __restrict__ rp,  const float* __restrict__ ry,
                 const float* __restrict__ w1,  const float* __restrict__ b1,
                 const float* __restrict__ w2,  const float* __restrict__ b2,
                 float* __restrict__ out, int nTokens)
{
  __shared__ __attribute__((aligned(16))) float q_lds[WAVES_PB][TPW][QP];
  __shared__ __attribute__((aligned(16))) float h_lds[WAVES_PB][TPW][QP];
  __shared__ __attribute__((aligned(16))) float f_lds[WAVES_PB][TPW][QP];
  __shared__ __attribute__((aligned(16))) float Hbuf [WAVES_PB][TPW][HP];

  const int wv    = threadIdx.x >> 5;
  const int lane  = threadIdx.x & 31;
  const int half  = lane >> 4;       // which 16-lane half of the wave
  const int l15   = lane & 15;
  const int khalf = half << 1;       // K sub-offset (0 or 2) inside a K=4 step
  const int tok0  = blockIdx.x * TPB + wv * TPW;

  // ---------------- Stage 1: closed-form circuit + LN1 + re-encode -----------
  // Lane t (< 16) owns token tok0+t entirely; LN is over 8 channels -> lane-local.
  if (lane < 16) {
    const int tok = tok0 + lane;
    float xv[NW];
    if (tok < nTokens) {
      const float4* xp = (const float4*)(x + (size_t)tok * NW);
      float4 v0 = xp[0], v1 = xp[1];
      xv[0]=v0.x; xv[1]=v0.y; xv[2]=v0.z; xv[3]=v0.w;
      xv[4]=v1.x; xv[5]=v1.y; xv[6]=v1.z; xv[7]=v1.w;
    } else {
      #pragma unroll
      for (int i = 0; i < NW; ++i) xv[i] = 0.f;
    }

    // <Z_w> after random_layer (RZ angles rp[2], rp[5] drop out analytically)
    const float cp1 = COSF(rp[1]);
    const float cp4 = COSF(rp[4]);
    float a[NW];
    a[0] = COSF(xv[0] + rp[0]);
    a[1] = a[0] * cp1 * COSF(xv[1]);
    const float c2 = COSF(xv[2]);
    a[2] = c2;
    a[3] = c2 * COSF(xv[3]);
    a[4] = COSF(xv[4] + rp[3]);
    a[5] = cp4 * COSF(xv[5]);
    a[6] = COSF(xv[6]);
    a[7] = COSF(xv[7]);

    // h = LayerNorm(x + a) * g1 + b1   (8-wide, lane-local)
    float hp[NW], mu = 0.f;
    #pragma unroll
    for (int i = 0; i < NW; ++i) { hp[i] = xv[i] + a[i]; mu += hp[i]; }
    mu *= 0.125f;
    float var = 0.f;
    #pragma unroll
    for (int i = 0; i < NW; ++i) { float d = hp[i] - mu; var += d * d; }
    var *= 0.125f;
    const float inv = rsqrtf(var + LN_EPS);
    #pragma unroll
    for (int i = 0; i < NW; ++i) {
      h_lds[wv][lane][i] = (hp[i] - mu) * inv * ln1g[i] + ln1b[i];
      // second circuit: RX(a) then RY(theta) -> <Z> = cos(theta)*cos(a)
      q_lds[wv][lane][i] = COSF(ry[i]) * COSF(a[i]);
    }
  }

  // ---------------- Stage 2: FFN via fp32 WMMA -------------------------------
  // GEMM1: C1(16x16) = Q(16x8) * W1^T(8x16-chunk), K=8 as 2 x K4 steps.
  const float* qrow = &q_lds[wv][l15][0];
  v2f aq0 = *(const v2f*)(qrow + 0 + khalf);   // K = 0..3 step
  v2f aq1 = *(const v2f*)(qrow + 4 + khalf);   // K = 4..7 step

  // GEMM2 B-operand setup: N columns 8..15 are padding. Instead of a divergent
  // guarded load (exec save/restore per K-step), always load from a valid w2
  // row (l15&7) and scale the fragment by a 0/1 lane mask -> branch-free loop.
  const float  msk   = (l15 < 8) ? 1.f : 0.f;
  const float* w2row = w2 + (size_t)(l15 & 7) * FFN_DIM + khalf;

  v8f acc2 = {};                               // GEMM2 accumulator (16 tok x 16, cols 0..7 valid)

  #pragma unroll 4
  for (int nc = 0; nc < FFN_DIM / 16; ++nc) {
    const int ncol = nc * 16 + l15;
    // B frag for GEMM1: B(k,n) = w1[n][k]; w1 row-major (1024,8) -> contiguous v2f
    const float* w1p = w1 + (size_t)ncol * NW + khalf;
    v2f bw0 = *(const v2f*)(w1p);
    v2f bw1 = *(const v2f*)(w1p + 4);

    v8f c1 = {};
    c1 = wmma4(aq0, bw0, c1);
    c1 = wmma4(aq1, bw1, c1);

    // bias + relu, park hidden chunk in LDS (C-layout -> A-layout conversion)
    const float bias = b1[ncol];
    #pragma unroll
    for (int r = 0; r < 8; ++r) {
      float hv = c1[r] + bias;
      Hbuf[wv][r + 8 * half][l15] = hv > 0.f ? hv : 0.f;
    }

    // GEMM2 partial: A = relu(H) chunk (16x16), B(k,o) = w2[o][k], K=16 as 4 x K4
    const float* hrow = &Hbuf[wv][l15][0];
    #pragma unroll
    for (int ks = 0; ks < 4; ++ks) {
      const int kloc = ks * 4 + khalf;
      v2f a2 = *(const v2f*)(hrow + kloc);
      v2f bw = *(const v2f*)(w2row + nc * 16 + ks * 4);
      bw.x *= msk;
      bw.y *= msk;
      acc2 = wmma4(a2, bw, acc2);
    }
  }

  // C2 fragment -> LDS (only the 8 valid output columns)
  if (l15 < 8) {
    #pragma unroll
    for (int r = 0; r < 8; ++r)
      f_lds[wv][r + 8 * half][l15] = acc2[r];
  }

  // ---------------- Stage 3: residual + bias + LN2 ---------------------------
  if (lane < 16) {
    const int tok = tok0 + lane;
    if (tok < nTokens) {
      float y[NW], mu = 0.f;
      #pragma unroll
      for (int i = 0; i < NW; ++i) {
        y[i] = h_lds[wv][lane][i] + f_lds[wv][lane][i] + b2[i];
        mu += y[i];
      }
      mu *= 0.125f;
      float var = 0.f;
      #pragma unroll
      for (int i = 0; i < NW; ++i) { float d = y[i] - mu; var += d * d; }
      var *= 0.125f;
      const float inv = rsqrtf(var + LN_EPS);
      float o[NW];
      #pragma unroll
      for (int i = 0; i < NW; ++i)
        o[i] = (y[i] - mu) * inv * ln2g[i] + ln2b[i];
      float4* op = (float4*)(out + (size_t)tok * NW);
      op[0] = make_float4(o[0], o[1], o[2], o[3]);
      op[1] = make_float4(o[4], o[5], o[6], o[7]);
    }
  }
}

extern "C" void kernel_launch(void* const* d_in, const int* in_sizes, int n_in,
                              void* d_out, int out_size, void* d_ws, size_t ws_size,
                              hipStream_t stream) {
  (void)n_in; (void)out_size; (void)d_ws; (void)ws_size;
  const float* x    = (const float*)d_in[0];
  const float* ln1g = (const float*)d_in[1];
  const float* ln1b = (const float*)d_in[2];
  const float* ln2g = (const float*)d_in[3];
  const float* ln2b = (const float*)d_in[4];
  const float* rp   = (const float*)d_in[5];
  const float* ry   = (const float*)d_in[6];
  const float* w1   = (const float*)d_in[7];
  const float* b1   = (const float*)d_in[8];
  const float* w2   = (const float*)d_in[9];
  const float* b2   = (const float*)d_in[10];

  const int nTokens = in_sizes[0] / NW;               // B*L = 131072
  const int blocks  = (nTokens + TPB - 1) / TPB;      // 1024 blocks of 256 threads

  qtb_fused_kernel<<<blocks, WAVES_PB * 32, 0, stream>>>(
      x, ln1g, ln1b, ln2g, ln2b, rp, ry, w1, b1, w2, b2,
      (float*)d_out, nTokens);
}